// ROIPoolingLayer_58995670777933
// MI455X (gfx1250) — compile-verified
//
#include <hip/hip_runtime.h>
#include <stdint.h>

// Feature map is fixed by the reference: (1, 256, 256, 256) NHWC f32.
constexpr int FM_W = 256;
constexpr int FM_C = 256;
constexpr int C4   = FM_C / 4;   // 64 float4 per pixel

// ---------------------------------------------------------------------------
// CDNA5 async copy: global -> LDS, 16B per lane, tracked by ASYNCcnt.
// vdst operand carries the LDS byte address (low 32 bits of the generic ptr).
// ---------------------------------------------------------------------------
__device__ __forceinline__ void async_gather_b128(uint32_t lds_addr,
                                                  const float* gaddr) {
    asm volatile("global_load_async_to_lds_b128 %0, %1, off"
                 :: "v"(lds_addr), "v"(gaddr)
                 : "memory");
}

struct RoiState {
    int   rx, ry, rw, rh;
    float sx, sy;          // w/P, h/P
};

__device__ __forceinline__ void load_roi(const int* __restrict__ rois, int roi,
                                         float fpool, RoiState& s) {
    s.rx = rois[roi * 5 + 1];
    s.ry = rois[roi * 5 + 2];
    s.rw = rois[roi * 5 + 3];
    s.rh = rois[roi * 5 + 4];
    s.sx = __fdiv_rn((float)s.rw, fpool);
    s.sy = __fdiv_rn((float)s.rh, fpool);
}

// One 64-thread group owns a contiguous chunk of (roi,py,px) cells; each
// thread owns one float4 of channels. Corner gathers for the next cell are
// double-buffered through LDS via the gfx1250 async copy engine while the
// current cell is blended.
__global__ __launch_bounds__(256) void roi_pool_async_kernel(
    const float* __restrict__ fm, const int* __restrict__ rois,
    float* __restrict__ out, int total, int pool, int chunk)
{
    __shared__ float4 stage[4][2][4][C4];   // [group][buf][corner][lane4] = 32 KB

    const int g     = threadIdx.x >> 6;
    const int lane4 = threadIdx.x & 63;
    const int pp    = pool * pool;
    const float fpool = (float)pool;

    const int group = blockIdx.x * 4 + g;
    const int base  = group * chunk;
    const int end   = min(base + chunk, total);
    if (base >= end) return;

    // One-time decomposition of the chunk start (32-bit divisions).
    int cell = base;
    int roi  = cell / pp;
    int rem  = cell - roi * pp;
    int wpy  = rem / pool;
    int wpx  = rem - wpy * pool;

    RoiState rs;
    load_roi(rois, roi, fpool, rs);

    const int coff = lane4 * 4;

    // Compute sample coords for (rs, py, px) — matches jnp:
    // f=(j+.5)*(len/P)-.5 clipped to [0,len-1]; *_rn ops forbid FMA
    // contraction so floor boundaries agree — then kick off the 4 corner
    // gathers into stage[g][st].
    auto issue = [&](int py, int px, int st, float& fx_out, float& fy_out) {
        float fxv = __fsub_rn(__fmul_rn((float)px + 0.5f, rs.sx), 0.5f);
        float fyv = __fsub_rn(__fmul_rn((float)py + 0.5f, rs.sy), 0.5f);
        fxv = fminf(fmaxf(fxv, 0.0f), (float)rs.rw - 1.0f);
        fyv = fminf(fmaxf(fyv, 0.0f), (float)rs.rh - 1.0f);
        int ix0 = (int)floorf(fxv);
        int iy0 = (int)floorf(fyv);
        int ix1 = min(ix0 + 1, rs.rw - 1);
        int iy1 = min(iy0 + 1, rs.rh - 1);
        fx_out = fxv - (float)ix0;
        fy_out = fyv - (float)iy0;
        int x0 = rs.rx + ix0, x1 = rs.rx + ix1;
        int y0 = rs.ry + iy0, y1 = rs.ry + iy1;

        const float* p00 = fm + (size_t)(y0 * FM_W + x0) * FM_C + coff;
        const float* p01 = fm + (size_t)(y0 * FM_W + x1) * FM_C + coff;
        const float* p10 = fm + (size_t)(y1 * FM_W + x0) * FM_C + coff;
        const float* p11 = fm + (size_t)(y1 * FM_W + x1) * FM_C + coff;

        async_gather_b128((uint32_t)(uintptr_t)&stage[g][st][0][lane4], p00);
        async_gather_b128((uint32_t)(uintptr_t)&stage[g][st][1][lane4], p01);
        async_gather_b128((uint32_t)(uintptr_t)&stage[g][st][2][lane4], p10);
        async_gather_b128((uint32_t)(uintptr_t)&stage[g][st][3][lane4], p11);
    };

    int st = 0;
    float cfx, cfy;
    issue(wpy, wpx, 0, cfx, cfy);

    while (cell < end) {
        const int next = cell + 1;
        float nfx = 0.0f, nfy = 0.0f;
        if (next < end) {
            // Incremental (roi,py,px) walk — no divisions in the loop.
            if (++wpx == pool) {
                wpx = 0;
                if (++wpy == pool) {
                    wpy = 0;
                    ++roi;
                    load_roi(rois, roi, fpool, rs);
                }
            }
            issue(wpy, wpx, st ^ 1, nfx, nfy);
            // 4 newer gathers outstanding; previous stage complete (in-order).
            asm volatile("s_wait_asynccnt 0x4" ::: "memory");
        } else {
            asm volatile("s_wait_asynccnt 0x0" ::: "memory");
        }

        float4 v00 = stage[g][st][0][lane4];
        float4 v01 = stage[g][st][1][lane4];
        float4 v10 = stage[g][st][2][lane4];
        float4 v11 = stage[g][st][3][lane4];

        const float wx1 = cfx, wx0 = 1.0f - cfx;
        const float wy1 = cfy, wy0 = 1.0f - cfy;
        float4 o;
        o.x = (v00.x * wx0 + v01.x * wx1) * wy0 + (v10.x * wx0 + v11.x * wx1) * wy1;
        o.y = (v00.y * wx0 + v01.y * wx1) * wy0 + (v10.y * wx0 + v11.y * wx1) * wy1;
        o.z = (v00.z * wx0 + v01.z * wx1) * wy0 + (v10.z * wx0 + v11.z * wx1) * wy1;
        o.w = (v00.w * wx0 + v01.w * wx1) * wy0 + (v10.w * wx0 + v11.w * wx1) * wy1;

        ((float4*)out)[(size_t)cell * C4 + lane4] = o;

        cell = next; st ^= 1; cfx = nfx; cfy = nfy;
    }
}

extern "C" void kernel_launch(void* const* d_in, const int* in_sizes, int n_in,
                              void* d_out, int out_size, void* d_ws, size_t ws_size,
                              hipStream_t stream) {
    (void)n_in; (void)d_ws; (void)ws_size;
    const float* fm   = (const float*)d_in[0];
    const int*   rois = (const int*)d_in[1];

    const int n_rois = in_sizes[1] / 5;
    const int pp     = out_size / (n_rois * FM_C);   // pool^2
    int pool = 1;
    while ((pool + 1) * (pool + 1) <= pp) ++pool;

    const int total = n_rois * pp;                   // cells (98,000)
    int blocks = (total + 3) / 4;                    // 4 groups per block
    if (blocks > 2048) blocks = 2048;
    if (blocks < 1) blocks = 1;
    const int groups = blocks * 4;
    const int chunk  = (total + groups - 1) / groups;

    roi_pool_async_kernel<<<blocks, 256, 0, stream>>>(
        fm, rois, (float*)d_out, total, pool, chunk);
}